// SketchDecoder_63376537419867
// MI455X (gfx1250) — compile-verified
//
#include <hip/hip_runtime.h>

#define HIDDEN 4096
#define MAXS   1000

typedef float v2f __attribute__((ext_vector_type(2)));
typedef float v8f __attribute__((ext_vector_type(8)));

#if __has_builtin(__builtin_amdgcn_wmma_f32_16x16x4_f32)
#define HAVE_WMMA 1
#else
#define HAVE_WMMA 0
#endif

__device__ __forceinline__ float fast_exp2(float x) {
    return __builtin_amdgcn_exp2f(x);
}

__device__ __forceinline__ float fast_tanh(float x) {
#if __has_builtin(__builtin_amdgcn_tanhf)
    return __builtin_amdgcn_tanhf(x);          // v_tanh_f32 (gfx1250 trans op)
#elif __has_builtin(__builtin_amdgcn_tanh_f32)
    return __builtin_amdgcn_tanh_f32(x);
#else
    // tanh(x) = 1 - 2/(2^(2*log2e*x) + 1); saturates correctly at +-inf
    float e = fast_exp2(x * 2.8853900817779268f);
    return 1.0f - 2.0f * __builtin_amdgcn_rcpf(e + 1.0f);
#endif
}

__device__ __forceinline__ float fast_sigmoid(float x) {
    return __builtin_fmaf(0.5f, fast_tanh(0.5f * x), 0.5f);
}

// One persistent workgroup: 1024 threads = 32 wave32s on one WGP.
// Wave w owns hidden units [128w, 128w+128). Lane layout inside a wave:
//   l16 = lane%16 selects the row within a 16-row WMMA tile,
//   half = lane/16 selects which 4 of the wave's 8 tiles this lane activates.
__global__ __launch_bounds__(1024) void sketch_decoder_kernel(
    const float* __restrict__ W_ih, const float* __restrict__ b_ih,
    const float* __restrict__ b_hh, const float* __restrict__ W_lin,
    const float* __restrict__ b_lin, const float* __restrict__ noise,
    float* __restrict__ out)
{
    __shared__ float s_wacc[32 * 8];  // per-wave logit partials (stride 8)
    __shared__ float s_prev[4];       // [x, y, p0, p1] (p2 folded out)
    __shared__ float s_blin[7];
    __shared__ int   s_act;

    const int tid  = threadIdx.x;
    const int wave = tid >> 5;
    const int lane = tid & 31;
    const int l16  = lane & 15;
    const int half = lane >> 4;
    const int hbase = wave * 128;

    // gate row offsets for i, g, o (f gate is dead: c_prev == 0)
    const int goff0 = 0, goff1 = 2 * HIDDEN, goff2 = 3 * HIDDEN;

#if HAVE_WMMA
    // Persistent WMMA B operands: B[k][n] = Wfold[rowbase+n][k], n = l16.
    // f32 4x16 B layout: v0 holds K=0 (lanes 0-15) / K=2 (lanes 16-31),
    //                    v1 holds K=1 / K=3.  Folded cols: [W0, W1, W2-W4, W3-W4].
    float B0[3][8], B1[3][8];
#else
    // Fallback: per-lane folded weights for the 4 hidden units it activates.
    float wf[3][4][4];
#endif
    float bias[3][4];   // folded bias = b_ih + b_hh + W[:,4]
    float wl[7][4];     // W_lin columns for this lane's hidden units

    {
        const int go[3] = {goff0, goff1, goff2};
#if HAVE_WMMA
        #pragma unroll
        for (int g = 0; g < 3; ++g) {
            #pragma unroll
            for (int t = 0; t < 8; ++t) {
                int row = go[g] + hbase + 16 * t + l16;
                const float* w = W_ih + row * 5;
                float w0 = w[0], w1 = w[1], w2 = w[2], w3 = w[3], w4 = w[4];
                B0[g][t] = half ? (w2 - w4) : w0;
                B1[g][t] = half ? (w3 - w4) : w1;
            }
        }
#endif
        #pragma unroll
        for (int ta = 0; ta < 4; ++ta) {
            int t = ta + 4 * half;
            int h = hbase + 16 * t + l16;
            #pragma unroll
            for (int g = 0; g < 3; ++g) {
                int row = go[g] + h;
                bias[g][ta] = b_ih[row] + b_hh[row] + W_ih[row * 5 + 4];
#if !HAVE_WMMA
                const float* w = W_ih + row * 5;
                wf[g][ta][0] = w[0];
                wf[g][ta][1] = w[1];
                wf[g][ta][2] = w[2] - w[4];
                wf[g][ta][3] = w[3] - w[4];
#endif
            }
            #pragma unroll
            for (int j = 0; j < 7; ++j) wl[j][ta] = W_lin[j * HIDDEN + h];
        }
    }

    if (tid == 0) {
        s_prev[0] = 0.f; s_prev[1] = 0.f; s_prev[2] = 1.f; s_prev[3] = 0.f;
        s_act = 1;
        out[0] = 0.f; out[1] = 0.f; out[2] = 1.f; out[3] = 0.f; out[4] = 0.f;
    }
    if (tid < 7) s_blin[tid] = b_lin[tid];
    __syncthreads();

    int t_step = 0;
    #pragma unroll 1
    for (; t_step < MAXS; ++t_step) {
        if (!s_act) break;  // uniform: frozen state => all remaining rows are 0
        const float px = s_prev[0], py = s_prev[1];
        const float pp0 = s_prev[2], pp1 = s_prev[3];

        float acc[7];
        #pragma unroll
        for (int j = 0; j < 7; ++j) acc[j] = 0.f;

#if HAVE_WMMA
        // A[m][k] = prev[k] for all m (16x4 f32 A layout: lanes 0-15 carry K=0,1;
        // lanes 16-31 carry K=2,3).
        v2f A;
        A.x = half ? pp0 : px;
        A.y = half ? pp1 : py;
        v8f cz = {};
#endif
        #pragma unroll
        for (int ta = 0; ta < 4; ++ta) {
            float iv, gv, ov;
#if HAVE_WMMA
            {
                v2f Blo, Bhi;
                Blo.x = B0[0][ta];     Blo.y = B1[0][ta];
                Bhi.x = B0[0][ta + 4]; Bhi.y = B1[0][ta + 4];
                v8f dlo = __builtin_amdgcn_wmma_f32_16x16x4_f32(false, A, false, Blo, (short)0, cz, false, false);
                v8f dhi = __builtin_amdgcn_wmma_f32_16x16x4_f32(false, A, false, Bhi, (short)0, cz, false, false);
                iv = (half ? dhi[0] : dlo[0]) + bias[0][ta];
            }
            {
                v2f Blo, Bhi;
                Blo.x = B0[1][ta];     Blo.y = B1[1][ta];
                Bhi.x = B0[1][ta + 4]; Bhi.y = B1[1][ta + 4];
                v8f dlo = __builtin_amdgcn_wmma_f32_16x16x4_f32(false, A, false, Blo, (short)0, cz, false, false);
                v8f dhi = __builtin_amdgcn_wmma_f32_16x16x4_f32(false, A, false, Bhi, (short)0, cz, false, false);
                gv = (half ? dhi[0] : dlo[0]) + bias[1][ta];
            }
            {
                v2f Blo, Bhi;
                Blo.x = B0[2][ta];     Blo.y = B1[2][ta];
                Bhi.x = B0[2][ta + 4]; Bhi.y = B1[2][ta + 4];
                v8f dlo = __builtin_amdgcn_wmma_f32_16x16x4_f32(false, A, false, Blo, (short)0, cz, false, false);
                v8f dhi = __builtin_amdgcn_wmma_f32_16x16x4_f32(false, A, false, Bhi, (short)0, cz, false, false);
                ov = (half ? dhi[0] : dlo[0]) + bias[2][ta];
            }
#else
            iv = bias[0][ta] + px * wf[0][ta][0] + py * wf[0][ta][1] + pp0 * wf[0][ta][2] + pp1 * wf[0][ta][3];
            gv = bias[1][ta] + px * wf[1][ta][0] + py * wf[1][ta][1] + pp0 * wf[1][ta][2] + pp1 * wf[1][ta][3];
            ov = bias[2][ta] + px * wf[2][ta][0] + py * wf[2][ta][1] + pp0 * wf[2][ta][2] + pp1 * wf[2][ta][3];
#endif
            float cv = fast_sigmoid(iv) * fast_tanh(gv);
            float hv = fast_sigmoid(ov) * fast_tanh(cv);
            #pragma unroll
            for (int j = 0; j < 7; ++j) acc[j] = __builtin_fmaf(wl[j][ta], hv, acc[j]);
        }

        // wave32 butterfly reduction of the 7 logit partials
        #pragma unroll
        for (int j = 0; j < 7; ++j) {
            float v = acc[j];
            v += __shfl_xor(v, 16, 32);
            v += __shfl_xor(v, 8, 32);
            v += __shfl_xor(v, 4, 32);
            v += __shfl_xor(v, 2, 32);
            v += __shfl_xor(v, 1, 32);
            acc[j] = v;
        }
        if (lane == 0) {
            #pragma unroll
            for (int j = 0; j < 7; ++j) s_wacc[wave * 8 + j] = acc[j];
        }
        __syncthreads();

        if (wave == 0) {
            float s[7];
            #pragma unroll
            for (int j = 0; j < 7; ++j) s[j] = s_wacc[lane * 8 + j];
            #pragma unroll
            for (int j = 0; j < 7; ++j) {
                float v = s[j];
                v += __shfl_xor(v, 16, 32);
                v += __shfl_xor(v, 8, 32);
                v += __shfl_xor(v, 4, 32);
                v += __shfl_xor(v, 2, 32);
                v += __shfl_xor(v, 1, 32);
                s[j] = v;
            }
            if (lane == 0) {
                const float L2E = 1.4426950408889634f;
                float l[7];
                #pragma unroll
                for (int j = 0; j < 7; ++j) l[j] = fast_tanh(s[j] + s_blin[j]);
                // softmax over l[4..6]
                float m  = fmaxf(l[4], fmaxf(l[5], l[6]));
                float e0 = fast_exp2((l[4] - m) * L2E);
                float e1 = fast_exp2((l[5] - m) * L2E);
                float e2 = fast_exp2((l[6] - m) * L2E);
                float inv = 1.0f / (e0 + e1 + e2);
                float q0 = e0 * inv, q1 = e1 * inv, q2 = e2 * inv;
                float nx = noise[2 * t_step], ny = noise[2 * t_step + 1];
                float x = l[0] + fast_exp2(l[1] * (0.5f * L2E)) * nx;
                float y = l[2] + fast_exp2(l[3] * (0.5f * L2E)) * ny;
                float* orow = out + 5 * (t_step + 1);
                orow[0] = x; orow[1] = y; orow[2] = q0; orow[3] = q1; orow[4] = q2;
                s_prev[0] = x; s_prev[1] = y; s_prev[2] = q0; s_prev[3] = q1;
                // argmax([q0,q1,q2]) == 2 iff q2 strictly beats both (ties -> earlier idx)
                s_act = !((q2 > q0) && (q2 > q1));
            }
        }
        __syncthreads();
    }

    // Rows after deactivation are exactly zero in the reference.
    for (int idx = 5 * (t_step + 1) + tid; idx < 5 * (MAXS + 1); idx += 1024)
        out[idx] = 0.f;
}

extern "C" void kernel_launch(void* const* d_in, const int* in_sizes, int n_in,
                              void* d_out, int out_size, void* d_ws, size_t ws_size,
                              hipStream_t stream) {
    const float* W_ih  = (const float*)d_in[0];
    // d_in[1] = W_hh: provably unused (h = c = 0 every step in the reference)
    const float* b_ih  = (const float*)d_in[2];
    const float* b_hh  = (const float*)d_in[3];
    const float* W_lin = (const float*)d_in[4];
    const float* b_lin = (const float*)d_in[5];
    const float* noise = (const float*)d_in[6];
    float* out = (float*)d_out;
    hipLaunchKernelGGL(sketch_decoder_kernel, dim3(1), dim3(1024), 0, stream,
                       W_ih, b_ih, b_hh, W_lin, b_lin, noise, out);
}